// Erosion_44186623541898
// MI455X (gfx1250) — compile-verified
//
#include <hip/hip_runtime.h>
#include <hip/hip_bf16.h>
#include <stdint.h>

#define MAXV 10000.0f

typedef float v4f __attribute__((ext_vector_type(4)));

static constexpr int W_     = 1024;
static constexpr int H_     = 1024;
static constexpr int TILE_H = 16;              // output rows per block
static constexpr int LROWS  = TILE_H + 2;      // staged rows (with halo) = 18
static constexpr int TPB    = W_ / 4;          // 256 threads: one float4 per thread per row

__device__ __forceinline__ v4f vmin3(v4f a, v4f b, v4f c) {
    v4f r;
    r.x = fminf(fminf(a.x, b.x), c.x);
    r.y = fminf(fminf(a.y, b.y), c.y);
    r.z = fminf(fminf(a.z, b.z), c.z);
    r.w = fminf(fminf(a.w, b.w), c.w);
    return r;
}

// Async-stage one row's float4 (per lane) into LDS, streaming (non-temporal).
__device__ __forceinline__ void async_row(const float* gp, float* lp) {
    const unsigned ldsOff = (unsigned)(uintptr_t)lp;  // low 32 bits of generic LDS ptr
    asm volatile("global_load_async_to_lds_b128 %0, %1, off th:TH_LOAD_NT"
                 :: "v"(ldsOff), "v"(gp)
                 : "memory");
}

__global__ void erode3x3_kernel(const float* __restrict__ x,
                                const float* __restrict__ kern,
                                float* __restrict__ out) {
    extern __shared__ float smem[];            // LROWS * W_ floats (dynamic LDS, 72 KB)

    const int  t     = threadIdx.x;            // 0..255
    const int  c     = t * 4;                  // first column owned by this thread
    const int  row0  = blockIdx.x * TILE_H;    // first output row of this tile
    const long long pbase = (long long)blockIdx.y * (long long)H_ * W_;
    const float* plane = x + pbase;

    // Structuring-element biases, exactly as reference: neigh = (k==0 ? -MAXV : 0),
    // contribution = value - neigh.
    float n[3][3];
    bool dense = true;
#pragma unroll
    for (int i = 0; i < 3; ++i) {
#pragma unroll
        for (int j = 0; j < 3; ++j) {
            const float kv = kern[i * 3 + j];
            n[i][j] = (kv == 0.0f) ? -MAXV : 0.0f;
            dense = dense && (kv != 0.0f);
        }
    }

    // ---- Phase 1: stage LROWS rows into LDS via gfx1250 async copies --------
    // Only slot 0 (row0-1) and slot LROWS-1 (row0+TILE_H) can be out of image;
    // the TILE_H interior rows are unconditionally in-bounds -> straight-line
    // burst of async loads, halo handling peeled into two uniform branches.
    const v4f pad = {MAXV, MAXV, MAXV, MAXV};

    if (row0 == 0) {
        *(v4f*)&smem[c] = pad;                                   // geodesic top border
    } else {
        async_row(plane + (long long)(row0 - 1) * W_ + c, &smem[c]);
    }
#pragma unroll
    for (int s = 1; s <= TILE_H; ++s) {                          // rows row0 .. row0+TILE_H-1
        async_row(plane + (long long)(row0 - 1 + s) * W_ + c, &smem[s * W_ + c]);
    }
    if (row0 + TILE_H >= H_) {
        *(v4f*)&smem[(LROWS - 1) * W_ + c] = pad;                // geodesic bottom border
    } else {
        async_row(plane + (long long)(row0 + TILE_H) * W_ + c, &smem[(LROWS - 1) * W_ + c]);
    }

    asm volatile("s_wait_asynccnt 0x0" ::: "memory");
    __syncthreads();

    // ---- Phase 2: horizontal mins per row, rolling vertical min ------------
    const bool le = (t == 0);
    const bool re = (c + 4 >= W_);

    // hmin of LDS slot s for kernel-row i: h[i] = min_j( in[c+l-1+j] - n[i][j] )
    auto hrow = [&](int s, v4f h[3]) {
        const float* rowp = &smem[s * W_];
        const v4f   v   = *(const v4f*)(rowp + c);
        const float lft = le ? MAXV : rowp[c - 1];
        const float rgt = re ? MAXV : rowp[c + 4];
        v4f w0 = {lft, v.x, v.y, v.z};         // in[c+l-1]
        v4f w1 = v;                            // in[c+l]
        v4f w2 = {v.y, v.z, v.w, rgt};         // in[c+l+1]
        if (dense) {
            h[0] = vmin3(w0, w1, w2);
            h[1] = h[0];
            h[2] = h[0];
        } else {
#pragma unroll
            for (int i = 0; i < 3; ++i) {
                h[i] = vmin3(w0 - n[i][0], w1 - n[i][1], w2 - n[i][2]);
            }
        }
    };

    // Peel slots 0 and 1 to prime the rolling registers.
    v4f h[3], h0pp, h0p, h1p;
    hrow(0, h);  h0pp = h[0];
    hrow(1, h);  h0p  = h[0];  h1p = h[1];

    for (int s = 2; s < LROWS; ++s) {
        hrow(s, h);
        // out[row0+s-2] = min( h0(slot s-2), h1(slot s-1), h2(slot s) )
        v4f r = vmin3(h0pp, h1p, h[2]);
        __builtin_nontemporal_store(
            r, (v4f*)(out + pbase + (long long)(row0 + s - 2) * W_ + c));
        h0pp = h0p;
        h0p  = h[0];
        h1p  = h[1];
    }
}

extern "C" void kernel_launch(void* const* d_in, const int* in_sizes, int n_in,
                              void* d_out, int out_size, void* d_ws, size_t ws_size,
                              hipStream_t stream) {
    const float* x    = (const float*)d_in[0];
    const float* kern = (const float*)d_in[1];
    float*       out  = (float*)d_out;

    const int planes = in_sizes[0] / (H_ * W_);        // B*C = 64
    dim3 grid(H_ / TILE_H, planes, 1);                 // 64 x 64 blocks
    dim3 block(TPB, 1, 1);                             // 256 threads = 8 waves (wave32)
    const size_t shmem = (size_t)LROWS * W_ * sizeof(float);  // 72 KB -> 4 blocks/WGP

    (void)hipFuncSetAttribute((const void*)erode3x3_kernel,
                              hipFuncAttributeMaxDynamicSharedMemorySize,
                              (int)shmem);
    erode3x3_kernel<<<grid, block, shmem, stream>>>(x, kern, out);

    (void)n_in; (void)out_size; (void)d_ws; (void)ws_size;
}